// GCN_48919677501959
// MI455X (gfx1250) — compile-verified
//
#include <hip/hip_runtime.h>
#include <hip/hip_bf16.h>

typedef __attribute__((ext_vector_type(2))) float v2f;
typedef __attribute__((ext_vector_type(8))) float v8f;

#define TB 256

// ---------------------------------------------------------------- degree / norm
__global__ void GCN_deg_init(float* __restrict__ deg, int N) {
    int i = blockIdx.x * TB + threadIdx.x;
    if (i < N) deg[i] = 1.0f;               // self loop
}

__global__ void GCN_deg_count(const int* __restrict__ col, float* __restrict__ deg, int E) {
    int e = blockIdx.x * TB + threadIdx.x;
    if (e < E) atomicAdd(&deg[col[e]], 1.0f);
}

__global__ void GCN_rsqrt(float* __restrict__ d, int N) {
    int i = blockIdx.x * TB + threadIdx.x;
    if (i < N) d[i] = rsqrtf(d[i]);          // deg >= 1 always
}

__global__ void GCN_norm(const int* __restrict__ row, const int* __restrict__ col,
                         const float* __restrict__ dinv, float* __restrict__ nrm, int E) {
    int e = blockIdx.x * TB + threadIdx.x;
    if (e < E) nrm[e] = dinv[row[e]] * dinv[col[e]];
}

// ---------------------------------------------------------------- weight pre-pack
// Pack W[K=128][ncols_valid] (row-major) into K-pair-contiguous, zero-padded form:
//   Wp[(p*NPAD + n)*2 + v] = (n < ncols_valid) ? W[(2p+v)*ncols_valid + n] : 0
// so each GEMM lane reads its two K values for column n as one b64 load.
__global__ void GCN_pack_w(const float* __restrict__ W, float* __restrict__ Wp,
                           int ncols_valid, int npad) {
    int i = blockIdx.x * TB + threadIdx.x;       // over 64 * npad * 2
    int total = 64 * npad * 2;
    if (i >= total) return;
    int p   = i / (npad * 2);
    int rem = i - p * (npad * 2);
    int n   = rem >> 1;
    int v   = rem & 1;
    Wp[i] = (n < ncols_valid) ? W[(size_t)(2 * p + v) * ncols_valid + n] : 0.0f;
}

// ---------------------------------------------------------------- fp32 WMMA GEMM
// D[M, NCOLS] = A[M,128] @ Wp (packed, NPAD columns, NPAD >= NCOLS), M multiple of 16.
// One wave per 16x16 output tile; branch-free inner loop: b64 + b64 + v_wmma per K-step.
template <int NPAD, int NCOLS>
__global__ void GCN_gemm_f32_wmma(const float* __restrict__ A, const float* __restrict__ Bp,
                                  float* __restrict__ D, int mTiles, int nTiles) {
    const int wave = threadIdx.x >> 5;
    const int lane = threadIdx.x & 31;
    const int half = lane >> 4;       // K split: lanes 0-15 -> K pair 0, lanes 16-31 -> K pair 1
    const int ln   = lane & 15;       // M index (A) / N index (B)

    const int tile = blockIdx.x * 8 + wave;          // wave-uniform
    if (tile >= mTiles * nTiles) return;             // wave-uniform exit: EXEC stays all-1s
    const int mT = tile / nTiles;
    const int nT = tile % nTiles;

    const float* __restrict__ arow = A + (size_t)(mT * 16 + ln) * 128 + 2 * half;
    const int n = nT * 16 + ln;                      // padded column index (< NPAD)
    const float* __restrict__ bcol = Bp + (size_t)n * 2 + half * (NPAD * 2);

    v8f c = {};
    #pragma unroll
    for (int kk = 0; kk < 128; kk += 4) {
        // A 16x4 f32 layout: VGPR0 = K (0|2), VGPR1 = K (1|3) by lane half
        v2f a = *(const v2f*)(arow + kk);
        // packed B: pair p = kk/2 + half, contiguous {K=2p, K=2p+1} for column n
        v2f b = *(const v2f*)(bcol + (size_t)(kk >> 1) * (NPAD * 2));
        c = __builtin_amdgcn_wmma_f32_16x16x4_f32(false, a, false, b, (short)0, c, false, false);
    }

    if (n < NCOLS) {                                 // only divergence: final store guard
        // C/D layout: VGPR r -> M = r + 8*half, N = lane&15
        float* dp = D + (size_t)(mT * 16 + half * 8) * NCOLS + n;
        #pragma unroll
        for (int r = 0; r < 8; ++r) dp[(size_t)r * NCOLS] = c[r];
    }
}

// ---------------------------------------------------------------- bias init / relu
__global__ void GCN_bias_init(float* __restrict__ buf, const float* __restrict__ b,
                              int total, int F) {
    int i = blockIdx.x * TB + threadIdx.x;
    if (i < total) buf[i] = b[i % F];
}

__global__ void GCN_relu(float* __restrict__ buf, int total) {
    int i = blockIdx.x * TB + threadIdx.x;
    if (i < total) buf[i] = fmaxf(buf[i], 0.0f);
}

// ---------------------------------------------------------------- edge scatter (128 feats)
// One warp per edge; 32 float4 chunks. Self loops are the tail range [E, E+N).
__global__ void GCN_scatter128(const int* __restrict__ row, const int* __restrict__ col,
                               const float* __restrict__ nrm, const float* __restrict__ dinv,
                               const float* __restrict__ src, float* __restrict__ dst,
                               int E, int N) {
    unsigned tid = blockIdx.x * TB + threadIdx.x;
    unsigned e = tid >> 5, ch = tid & 31;
    if (e >= (unsigned)(E + N)) return;
    int r, c; float w;
    if (e < (unsigned)E) { r = row[e]; c = col[e]; w = nrm[e]; }
    else { int i = (int)e - E; r = i; c = i; float dv = dinv[i]; w = dv * dv; }
    const float4 v = *(const float4*)(src + (size_t)r * 128 + ch * 4);
    float* d = dst + (size_t)c * 128 + ch * 4;
    atomicAdd(d + 0, v.x * w);
    atomicAdd(d + 1, v.y * w);
    atomicAdd(d + 2, v.z * w);
    atomicAdd(d + 3, v.w * w);
}

// ---------------------------------------------------------------- edge scatter (40 feats)
// Half a warp (16 lanes) per edge; 16 chunk slots, 10 active (40 = 10 * float4).
__global__ void GCN_scatter40(const int* __restrict__ row, const int* __restrict__ col,
                              const float* __restrict__ nrm, const float* __restrict__ dinv,
                              const float* __restrict__ src, float* __restrict__ dst,
                              int E, int N) {
    unsigned tid = blockIdx.x * TB + threadIdx.x;
    unsigned e = tid >> 4, ch = tid & 15;
    if (e >= (unsigned)(E + N) || ch >= 10u) return;
    int r, c; float w;
    if (e < (unsigned)E) { r = row[e]; c = col[e]; w = nrm[e]; }
    else { int i = (int)e - E; r = i; c = i; float dv = dinv[i]; w = dv * dv; }
    const float4 v = *(const float4*)(src + (size_t)r * 40 + ch * 4);
    float* d = dst + (size_t)c * 40 + ch * 4;
    atomicAdd(d + 0, v.x * w);
    atomicAdd(d + 1, v.y * w);
    atomicAdd(d + 2, v.z * w);
    atomicAdd(d + 3, v.w * w);
}

// ---------------------------------------------------------------- launch
extern "C" void kernel_launch(void* const* d_in, const int* in_sizes, int n_in,
                              void* d_out, int out_size, void* d_ws, size_t ws_size,
                              hipStream_t stream) {
    const float* X  = (const float*)d_in[0];
    const int*   EI = (const int*)d_in[1];
    const float* W1 = (const float*)d_in[2];
    const float* b1 = (const float*)d_in[3];
    const float* W2 = (const float*)d_in[4];
    const float* b2 = (const float*)d_in[5];
    float* out = (float*)d_out;

    const int N = in_sizes[0] / 128;     // 50000 (multiple of 16)
    const int E = in_sizes[1] / 2;       // 800000
    const int* row = EI;                 // source
    const int* col = EI + E;             // target

    // workspace: dinv[N] | norm[E] | T1[N*128] | AGG[N*128] | Wp1[64*128*2] | Wp2[64*48*2]
    float* dinv = (float*)d_ws;
    float* nrm  = dinv + N;
    float* T1   = nrm + E;
    float* AGG  = T1 + (size_t)N * 128;
    float* Wp1  = AGG + (size_t)N * 128;
    float* Wp2  = Wp1 + 64 * 128 * 2;
    float* T2   = T1;                    // reuse

    const int gN = (N + TB - 1) / TB;
    const int gE = (E + TB - 1) / TB;
    const int mT = N / 16;               // 3125

    // degrees / normalization
    GCN_deg_init <<<gN, TB, 0, stream>>>(dinv, N);
    GCN_deg_count<<<gE, TB, 0, stream>>>(col, dinv, E);
    GCN_rsqrt    <<<gN, TB, 0, stream>>>(dinv, N);
    GCN_norm     <<<gE, TB, 0, stream>>>(row, col, dinv, nrm, E);

    // pack weights into WMMA-friendly K-pair layout (zero-padded to 48 for W2)
    GCN_pack_w<<<(64 * 128 * 2 + TB - 1) / TB, TB, 0, stream>>>(W1, Wp1, 128, 128);
    GCN_pack_w<<<(64 *  48 * 2 + TB - 1) / TB, TB, 0, stream>>>(W2, Wp2,  40,  48);

    // layer 1: T1 = X @ W1  (128 cols -> 8 N-tiles)
    {
        int tiles = mT * 8;
        GCN_gemm_f32_wmma<128, 128><<<(tiles + 7) / 8, TB, 0, stream>>>(X, Wp1, T1, mT, 8);
    }
    // AGG = b1 broadcast; scatter messages; relu
    {
        int total = N * 128;
        GCN_bias_init<<<(total + TB - 1) / TB, TB, 0, stream>>>(AGG, b1, total, 128);
        unsigned thr = (unsigned)(E + N) * 32u;
        GCN_scatter128<<<(thr + TB - 1) / TB, TB, 0, stream>>>(row, col, nrm, dinv, T1, AGG, E, N);
        GCN_relu<<<(total + TB - 1) / TB, TB, 0, stream>>>(AGG, total);
    }

    // layer 2: T2 = AGG @ W2  (48 padded cols -> 3 N-tiles, store-guarded to 40)
    {
        int tiles = mT * 3;
        GCN_gemm_f32_wmma<48, 40><<<(tiles + 7) / 8, TB, 0, stream>>>(AGG, Wp2, T2, mT, 3);
    }
    // out = b2 broadcast; scatter messages
    {
        int total = N * 40;
        GCN_bias_init<<<(total + TB - 1) / TB, TB, 0, stream>>>(out, b2, total, 40);
        unsigned thr = (unsigned)(E + N) * 16u;
        GCN_scatter40<<<(thr + TB - 1) / TB, TB, 0, stream>>>(row, col, nrm, dinv, T2, out, E, N);
    }
}